// VGG_GCN_graph_15221364097535
// MI455X (gfx1250) — compile-verified
//
#include <hip/hip_runtime.h>
#include <math.h>

#define BN 4096
#define DD 512
#define NC 10
#define KTOP 5

typedef float v2f __attribute__((ext_vector_type(2)));
typedef float v8f __attribute__((ext_vector_type(8)));

static __device__ __forceinline__ v8f wmma4(v2f a, v2f b, v8f c) {
    // D = A(16x4, f32) * B(4x16, f32) + C(16x16, f32)
    return __builtin_amdgcn_wmma_f32_16x16x4_f32(
        /*neg_a=*/false, a, /*neg_b=*/false, b,
        /*c_mod=*/(short)0, c, /*reuse_a=*/false, /*reuse_b=*/false);
}

// ---------------------------------------------------------------------------
// Y[M,N] = act(X[M,DD] @ W[DD,N] + bias); one 16x16 tile per wave, 4 waves/block
// A slab (16 rows x 512) staged in LDS, pitch 516 to avoid bank conflicts.
// ---------------------------------------------------------------------------
__global__ __launch_bounds__(128) void linear_wmma(
    const float* __restrict__ X, const float* __restrict__ W,
    const float* __restrict__ bias, float* __restrict__ Y,
    int N, int do_relu) {
    __shared__ __align__(16) float ldsX[16 * 516];
    const int tid  = threadIdx.x;
    const int wave = tid >> 5, lane = tid & 31;
    const int rowBase = blockIdx.y << 4;
    const int colBase = (blockIdx.x << 6) + (wave << 4);

    for (int idx = tid; idx < 16 * 128; idx += 128) {
        int r = idx >> 7, c4 = (idx & 127) << 2;
        *(float4*)&ldsX[r * 516 + c4] = *(const float4*)&X[(size_t)(rowBase + r) * DD + c4];
    }
    __syncthreads();

    v8f acc = {};
    const int l15   = lane & 15;
    const int khalf = (lane >> 4) << 1;   // lanes 0-15: K k..k+1, lanes 16-31: K k+2..k+3
    const int col   = colBase + l15;
    #pragma unroll 4
    for (int k = 0; k < DD; k += 4) {
        int kk = k + khalf;
        v2f a = *(const v2f*)&ldsX[l15 * 516 + kk];
        v2f b;
        b.x = W[(size_t)kk * N + col];
        b.y = W[(size_t)(kk + 1) * N + col];
        acc = wmma4(a, b, acc);
    }
    float bv = bias[col];
    #pragma unroll
    for (int v = 0; v < 8; ++v) {
        int row = rowBase + v + ((lane >> 4) << 3);
        float val = acc[v] + bv;
        if (do_relu) val = fmaxf(val, 0.0f);
        Y[(size_t)row * N + col] = val;
    }
}

// ---------------------------------------------------------------------------
// graph_pre = sigmoid(low @ low^T). B = low^T, so B-frag is a float2 row load.
// ---------------------------------------------------------------------------
__global__ __launch_bounds__(128) void gram_wmma(
    const float* __restrict__ low, float* __restrict__ gpre) {
    __shared__ __align__(16) float ldsA[16 * 516];
    const int tid  = threadIdx.x;
    const int wave = tid >> 5, lane = tid & 31;
    const int rowBase = blockIdx.y << 4;
    const int colBase = (blockIdx.x << 6) + (wave << 4);

    for (int idx = tid; idx < 16 * 128; idx += 128) {
        int r = idx >> 7, c4 = (idx & 127) << 2;
        *(float4*)&ldsA[r * 516 + c4] = *(const float4*)&low[(size_t)(rowBase + r) * DD + c4];
    }
    __syncthreads();

    v8f acc = {};
    const int l15   = lane & 15;
    const int khalf = (lane >> 4) << 1;
    const size_t jrow = (size_t)(colBase + l15) * DD;
    #pragma unroll 4
    for (int k = 0; k < DD; k += 4) {
        int kk = k + khalf;
        v2f a = *(const v2f*)&ldsA[l15 * 516 + kk];
        v2f b = *(const v2f*)&low[jrow + kk];   // B[k][n] = low[col_n][k]
        acc = wmma4(a, b, acc);
    }
    #pragma unroll
    for (int v = 0; v < 8; ++v) {
        int row = rowBase + v + ((lane >> 4) << 3);
        int col = colBase + l15;
        float s = 1.0f / (1.0f + __expf(-acc[v]));
        gpre[(size_t)row * BN + col] = s;
    }
}

// ---------------------------------------------------------------------------
// Per-row top-5 indices (lowest index wins ties, matching lax.top_k).
// Row cached in LDS; 5 masked argmax tree-reductions.
// ---------------------------------------------------------------------------
__global__ __launch_bounds__(256) void topk_kernel(
    const float* __restrict__ gpre, int* __restrict__ tki) {
    __shared__ __align__(16) float rowv[BN];
    __shared__ float rv[256];
    __shared__ int   ri[256];
    __shared__ int   ssel[KTOP];
    const int i = blockIdx.x, tid = threadIdx.x;
    const float* rp = gpre + (size_t)i * BN;

    for (int j4 = tid; j4 < BN / 4; j4 += 256)
        *(float4*)&rowv[j4 * 4] = *(const float4*)&rp[j4 * 4];
    __syncthreads();

    for (int t = 0; t < KTOP; ++t) {
        float best = -1.0f; int bidx = BN;
        for (int j = tid; j < BN; j += 256) {
            bool skip = false;
            for (int p = 0; p < t; ++p) skip = skip || (ssel[p] == j);
            if (skip) continue;
            float v = rowv[j];
            if (v > best) { best = v; bidx = j; }
        }
        rv[tid] = best; ri[tid] = bidx;
        __syncthreads();
        for (int off = 128; off; off >>= 1) {
            if (tid < off) {
                float v2 = rv[tid + off]; int i2 = ri[tid + off];
                if (v2 > rv[tid] || (v2 == rv[tid] && i2 < ri[tid])) {
                    rv[tid] = v2; ri[tid] = i2;
                }
            }
            __syncthreads();
        }
        if (tid == 0) ssel[t] = ri[0];
        __syncthreads();
    }
    if (tid < KTOP) tki[i * KTOP + tid] = ssel[tid];
}

// keep = mask | mask^T built by symmetric scatter (benign same-value races)
__global__ void scatter_keep(const int* __restrict__ tki,
                             unsigned char* __restrict__ keep) {
    int t = blockIdx.x * blockDim.x + threadIdx.x;
    if (t >= BN * KTOP) return;
    int i = t / KTOP;
    int j = tki[t];
    keep[(size_t)i * BN + j] = 1;
    keep[(size_t)j * BN + i] = 1;
}

// gp = (keep && gpre > 0.9); graph_gt = (tgt_i == tgt_j); dinv = 1/(rowsum(gp)+1)
__global__ __launch_bounds__(256) void gp_gt_rowsum(
    const float* __restrict__ gpre, const unsigned char* __restrict__ keep,
    const int* __restrict__ tgt, float* __restrict__ gp,
    float* __restrict__ gt, float* __restrict__ dinv) {
    __shared__ int scnt[256];
    const int i = blockIdx.x, tid = threadIdx.x;
    const int ti = tgt[i];
    int cnt = 0;
    for (int j = tid; j < BN; j += 256) {
        size_t idx = (size_t)i * BN + j;
        float g = gpre[idx];
        float gpv = (keep[idx] && g > 0.9f) ? 1.0f : 0.0f;
        gp[idx] = gpv;
        gt[idx] = (tgt[j] == ti) ? 1.0f : 0.0f;
        cnt += (gpv != 0.0f);
    }
    scnt[tid] = cnt;
    __syncthreads();
    for (int off = 128; off; off >>= 1) {
        if (tid < off) scnt[tid] += scnt[tid + off];
        __syncthreads();
    }
    if (tid == 0) dinv[i] = 1.0f / (float)(scnt[0] + 1);   // A_hat = gp + I
}

// probs = softmax(x @ Wc + bc); one wave per row (N=10 too small for WMMA)
__global__ __launch_bounds__(128) void classifier_softmax(
    const float* __restrict__ X, const float* __restrict__ Wc,
    const float* __restrict__ bc, float* __restrict__ probs) {
    __shared__ float lWc[DD * NC];
    const int tid = threadIdx.x;
    for (int idx = tid; idx < DD * NC; idx += 128) lWc[idx] = Wc[idx];
    __syncthreads();
    const int wave = tid >> 5, lane = tid & 31;
    const int r = (blockIdx.x << 2) + wave;
    float acc[NC];
    #pragma unroll
    for (int c = 0; c < NC; ++c) acc[c] = 0.0f;
    for (int d = lane; d < DD; d += 32) {
        float xv = X[(size_t)r * DD + d];
        #pragma unroll
        for (int c = 0; c < NC; ++c) acc[c] += xv * lWc[d * NC + c];
    }
    #pragma unroll
    for (int c = 0; c < NC; ++c)
        for (int off = 16; off; off >>= 1)
            acc[c] += __shfl_xor(acc[c], off, 32);
    if (lane == 0) {
        float m = -1e30f;
        #pragma unroll
        for (int c = 0; c < NC; ++c) { acc[c] += bc[c]; m = fmaxf(m, acc[c]); }
        float s = 0.0f;
        #pragma unroll
        for (int c = 0; c < NC; ++c) { acc[c] = __expf(acc[c] - m); s += acc[c]; }
        float inv = 1.0f / s;
        #pragma unroll
        for (int c = 0; c < NC; ++c) probs[r * NC + c] = acc[c] * inv;
    }
}

// out[i,c] = log( sum_j gp[i,j]*dinv[j]*probs[j,c] + dinv[i]*probs[i,c] )
// Deterministic serial reduction (no float atomics -> identical across replays).
__global__ __launch_bounds__(256) void aggregate_log(
    const float* __restrict__ gp, const float* __restrict__ probs,
    const float* __restrict__ dinv, float* __restrict__ out) {
    __shared__ float part[256][NC + 1];
    const int i = blockIdx.x, tid = threadIdx.x;
    float loc[NC];
    #pragma unroll
    for (int c = 0; c < NC; ++c) loc[c] = 0.0f;
    for (int j = tid; j < BN; j += 256) {
        float g = gp[(size_t)i * BN + j];
        if (g != 0.0f) {
            float dv = dinv[j];
            #pragma unroll
            for (int c = 0; c < NC; ++c) loc[c] += dv * probs[j * NC + c];
        }
    }
    #pragma unroll
    for (int c = 0; c < NC; ++c) part[tid][c] = loc[c];
    __syncthreads();
    if (tid < NC) {
        float s = 0.0f;
        for (int t = 0; t < 256; ++t) s += part[t][tid];
        s += dinv[i] * probs[i * NC + tid];
        out[i * NC + tid] = logf(s);
    }
}

// ---------------------------------------------------------------------------
extern "C" void kernel_launch(void* const* d_in, const int* in_sizes, int n_in,
                              void* d_out, int out_size, void* d_ws, size_t ws_size,
                              hipStream_t stream) {
    (void)in_sizes; (void)n_in; (void)out_size; (void)ws_size;
    const float* x   = (const float*)d_in[0];
    const int*   tgt = (const int*)  d_in[1];
    const float* W1  = (const float*)d_in[2];
    const float* b1  = (const float*)d_in[3];
    const float* W2  = (const float*)d_in[4];
    const float* b2  = (const float*)d_in[5];
    const float* Wc  = (const float*)d_in[6];
    const float* bc  = (const float*)d_in[7];

    // d_out layout (return order): out[B,C], graph_pre[B,B], graph_gt[B,B], gp[B,B]
    float* out  = (float*)d_out;
    float* gpre = out + (size_t)BN * NC;
    float* gt   = gpre + (size_t)BN * BN;
    float* gp   = gt   + (size_t)BN * BN;

    // workspace layout (~35 MB)
    char* w = (char*)d_ws;
    float*         h1    = (float*)w;                          // 8 MB
    float*         low   = (float*)(w + (size_t)(8u  << 20));  // 8 MB
    unsigned char* keep  = (unsigned char*)(w + (size_t)(16u << 20)); // 16 MB
    int*           tki   = (int*)  (w + (size_t)(32u << 20));  // 80 KB
    float*         dinv  = (float*)(w + (size_t)(33u << 20));  // 16 KB
    float*         probs = (float*)(w + (size_t)(34u << 20));  // 160 KB

    // low = relu(x@W1+b1) @ W2 + b2
    linear_wmma<<<dim3(DD / 64, BN / 16), 128, 0, stream>>>(x,  W1, b1, h1,  DD, 1);
    linear_wmma<<<dim3(DD / 64, BN / 16), 128, 0, stream>>>(h1, W2, b2, low, DD, 0);
    // graph_pre = sigmoid(low @ low^T)
    gram_wmma<<<dim3(BN / 64, BN / 16), 128, 0, stream>>>(low, gpre);
    // top-5 per row -> symmetric keep mask
    topk_kernel<<<BN, 256, 0, stream>>>(gpre, tki);
    hipMemsetAsync(keep, 0, (size_t)BN * BN, stream);
    scatter_keep<<<(BN * KTOP + 255) / 256, 256, 0, stream>>>(tki, keep);
    // gp / graph_gt / degree inverse
    gp_gt_rowsum<<<BN, 256, 0, stream>>>(gpre, keep, tgt, gp, gt, dinv);
    // classifier softmax, then normalized-adjacency aggregation + log
    classifier_softmax<<<BN / 4, 128, 0, stream>>>(x, Wc, bc, probs);
    aggregate_log<<<BN, 256, 0, stream>>>(gp, probs, dinv, out);
}